// MaxPool2D_65240553226693
// MI455X (gfx1250) — compile-verified
//
#include <hip/hip_runtime.h>

// 2x2 NHWC max-pool, f32.
// Input : (64, 224, 224, 96)  -> 308,281,344 floats (~1.233 GB)
// Output: (64, 112, 112, 96)  ->  77,070,336 floats (~308 MB)
// Pure HBM stream: ~66 us at 23.3 TB/s. Strategy: one float4 (4 channels) of
// one output pixel per thread; 4x global_load_b128 (TH=NT, zero reuse),
// 8 VALU max ops, 1x global_store_b128 (TH=NT). No divides anywhere.

typedef float v4f __attribute__((ext_vector_type(4)));

namespace {
constexpr int C     = 96;
constexpr int C4    = C / 4;        // 24 float4 per pixel
constexpr int IW    = 224;
constexpr int IH    = 224;
constexpr int OW    = 112;
constexpr int OH    = 112;
constexpr int BATCH = 64;
constexpr int OW_TILE = 8;          // output pixels per block along W

constexpr int IN_ROW4  = IW * C4;   // 5376 float4 per input row
constexpr int OUT_ROW4 = OW * C4;   // 2688 float4 per output row
}

__global__ __launch_bounds__(C4 * OW_TILE)  // 192 threads = 6 wave32
void maxpool2x2_nhwc_f32(const float* __restrict__ in, float* __restrict__ out) {
    const int c4 = threadIdx.x;                       // 0..23  (channel quad)
    const int ow = blockIdx.x * OW_TILE + threadIdx.y; // 0..111
    const int oh = blockIdx.y;                        // 0..111
    const int b  = blockIdx.z;                        // 0..63

    const v4f* __restrict__ in4  = reinterpret_cast<const v4f*>(in);
    v4f*       __restrict__ out4 = reinterpret_cast<v4f*>(out);

    // Base = input pixel (b, 2*oh, 2*ow), channel quad c4 (float4 units).
    const long long ibase =
        (static_cast<long long>(b) * IH + 2 * oh) * IN_ROW4 + (2 * ow) * C4 + c4;

    // Four pool-window pixels; neighbor offsets are compile-time constants
    // that fold into the load instruction's signed 24-bit ioffset:
    //   +C4       = +384 B   (w+1)
    //   +IN_ROW4  = +86016 B (h+1)
    //   +both     = +86400 B (h+1, w+1)
    const v4f p00 = __builtin_nontemporal_load(in4 + ibase);
    const v4f p01 = __builtin_nontemporal_load(in4 + ibase + C4);
    const v4f p10 = __builtin_nontemporal_load(in4 + ibase + IN_ROW4);
    const v4f p11 = __builtin_nontemporal_load(in4 + ibase + IN_ROW4 + C4);

    v4f m;
#pragma unroll
    for (int i = 0; i < 4; ++i) {
        // fmax tree; backend fuses toward v_max3_f32 / v_max_num_f32.
        m[i] = fmaxf(fmaxf(p00[i], p01[i]), fmaxf(p10[i], p11[i]));
    }

    const long long obase =
        (static_cast<long long>(b) * OH + oh) * OUT_ROW4 + ow * C4 + c4;
    __builtin_nontemporal_store(m, out4 + obase);
}

extern "C" void kernel_launch(void* const* d_in, const int* in_sizes, int n_in,
                              void* d_out, int out_size, void* d_ws, size_t ws_size,
                              hipStream_t stream) {
    (void)in_sizes; (void)n_in; (void)d_ws; (void)ws_size; (void)out_size;

    const float* in  = static_cast<const float*>(d_in[0]);
    float*       out = static_cast<float*>(d_out);

    // Exact cover, no bounds checks:
    //   block (24, 8)  : 24 channel-quads x 8 output pixels = 192 threads
    //   grid  (14,112,64): 14*8 = 112 output columns, 112 rows, 64 batches
    dim3 block(C4, OW_TILE, 1);
    dim3 grid(OW / OW_TILE, OH, BATCH);
    hipLaunchKernelGGL(maxpool2x2_nhwc_f32, grid, block, 0, stream, in, out);
}